// CreatePatches_51110110823149
// MI455X (gfx1250) — compile-verified
//
#include <hip/hip_runtime.h>

// CreatePatches: reflect-pad 4000x6000x3 -> 4096x6144x3, scale by 1/255,
// emit [384, 256, 256, 3] patches. Pure streaming op: ~590 MB total traffic,
// HBM-bound (~25 us floor at 23.3 TB/s). Strategy: B128 vector loads/stores
// with non-temporal (TH_NT) cache policy (working set >> 192 MB L2), scalar
// gather fallback only for the reflected rightmost patch column (~4% of data).

#define H   4000
#define W   6000
#define C   3
#define P   256
#define NH  16              // 4096 / 256
#define NW  24              // 6144 / 256
#define ROW_F   (P * C)     // 768 floats per patch row
#define ROW_F4  (ROW_F / 4) // 192 float4 per patch row
#define N_PROWS (NH * NW * P)   // 98304 patch rows total
#define ROWS_PER_BLOCK 4

typedef float v4f __attribute__((ext_vector_type(4)));

__global__ __launch_bounds__(ROW_F4 * ROWS_PER_BLOCK)
void create_patches_kernel(const float* __restrict__ in, float* __restrict__ out) {
    const unsigned f4     = threadIdx.x;                                   // 0..191
    const unsigned row_id = blockIdx.x * ROWS_PER_BLOCK + threadIdx.y;     // 0..98303
    if (row_id >= N_PROWS) return;

    const unsigned p_idx = row_id >> 8;          // patch index = row_id / 256
    const unsigned r     = row_id & 255u;        // row within patch
    const unsigned nh_i  = p_idx / NW;
    const unsigned nw_i  = p_idx - nh_i * NW;

    // Row reflection (bottom pad): src = 2*(H-1) - gr for gr >= H
    const int gr = (int)(nh_i * P + r);
    const int sr = (gr < H) ? gr : (2 * (H - 1) - gr);
    const float* __restrict__ inRow = in + (size_t)sr * (W * C);

    v4f v;
    if (nw_i != NW - 1) {
        // Interior patch columns: source segment contiguous & 16B-aligned.
        const v4f* __restrict__ src =
            (const v4f*)(inRow + (size_t)nw_i * ROW_F) + f4;
        v = __builtin_nontemporal_load(src);     // global_load_b128 th:NT
    } else {
        // Rightmost patch column: gc in [5888, 6143], reflect at gc >= W.
        const unsigned q = f4 * 4u;              // float offset within patch row
        float tmp[4];
#pragma unroll
        for (int j = 0; j < 4; ++j) {
            const unsigned qq = q + (unsigned)j;
            const unsigned cq = qq / 3u;         // column within patch
            const unsigned ch = qq - cq * 3u;    // channel
            const int gc = (int)((NW - 1) * P + cq);
            const int sc = (gc < W) ? gc : (2 * (W - 1) - gc);
            tmp[j] = inRow[(size_t)sc * C + ch];
        }
        v = (v4f){tmp[0], tmp[1], tmp[2], tmp[3]};
    }

    const float s = 1.0f / 255.0f;
    v *= s;

    // Output flat float4 index is exactly row_id*192 + f4.
    v4f* __restrict__ dst = (v4f*)out + ((size_t)row_id * ROW_F4 + f4);
    __builtin_nontemporal_store(v, dst);         // global_store_b128 th:NT
}

extern "C" void kernel_launch(void* const* d_in, const int* in_sizes, int n_in,
                              void* d_out, int out_size, void* d_ws, size_t ws_size,
                              hipStream_t stream) {
    (void)in_sizes; (void)n_in; (void)out_size; (void)d_ws; (void)ws_size;
    const float* in = (const float*)d_in[0];
    float* out = (float*)d_out;

    dim3 block(ROW_F4, ROWS_PER_BLOCK);                 // 192 x 4 = 768 threads (24 waves)
    dim3 grid(N_PROWS / ROWS_PER_BLOCK);                // 24576 blocks
    create_patches_kernel<<<grid, block, 0, stream>>>(in, out);
}